// GCNLayer_40535901340214
// MI455X (gfx1250) — compile-verified
//
#include <hip/hip_runtime.h>
#include <hip/hip_bf16.h>

typedef __attribute__((ext_vector_type(2))) float v2f;
typedef __attribute__((ext_vector_type(8))) float v8f;

#define IN_FEATS 128
#define OUT_FEATS 128

// ---------------------------------------------------------------------------
// Kernel 1: zero the aggregation workspace (float4-wide stores)
// ---------------------------------------------------------------------------
__global__ void gcn_zero_kernel(float4* __restrict__ p, long long n4) {
    long long i = (long long)blockIdx.x * blockDim.x + threadIdx.x;
    if (i < n4) p[i] = make_float4(0.f, 0.f, 0.f, 0.f);
}

// ---------------------------------------------------------------------------
// Kernel 2: edge aggregation. One wave32 per edge: each lane gathers a
// float4 of the 128-wide src row (coalesced 512B) and does 4 no-return
// global_atomic_add_f32 into agg[dst]. agg (51.2MB) stays L2-resident
// (192MB L2), so the atomic RMWs never leave the die.
// ---------------------------------------------------------------------------
__global__ void gcn_aggregate_kernel(const float* __restrict__ feat,
                                     const long long* __restrict__ esrc,
                                     const long long* __restrict__ edst,
                                     float* __restrict__ agg,
                                     long long n_edges) {
    const int lane = threadIdx.x & 31;
    long long edge = (long long)blockIdx.x * (blockDim.x >> 5) + (threadIdx.x >> 5);
    if (edge >= n_edges) return;

    const long long s = esrc[edge];
    const long long d = edst[edge];

    const float4 v = ((const float4*)(feat + s * IN_FEATS))[lane];
    float* dst = agg + d * IN_FEATS + lane * 4;
    atomicAdd(dst + 0, v.x);
    atomicAdd(dst + 1, v.y);
    atomicAdd(dst + 2, v.z);
    atomicAdd(dst + 3, v.w);
}

// ---------------------------------------------------------------------------
// Kernel 3: out = relu(agg @ W^T + b) via V_WMMA_F32_16X16X4_F32.
// Block = 256 threads = 8 waves. Block handles 16 nodes (M-tile); wave w
// handles output features [16w, 16w+16) (N-tile). K-loop: 32 WMMAs of K=4.
//
// A 16x4 f32 layout (ISA 7.12.2): lanes 0-15 -> M=lane, v0=K0, v1=K1;
//                                  lanes 16-31 -> M=lane-16, v0=K2, v1=K3.
// B 4x16 mirrored (N across lanes, K across VGPRs + lane halves).
// C/D: vgpr r, lanes 0-15 -> (M=r, N=lane); lanes 16-31 -> (M=r+8, N=lane-16).
// ---------------------------------------------------------------------------
__global__ void gcn_gemm_relu_kernel(const float* __restrict__ agg,
                                     const float* __restrict__ W,
                                     const float* __restrict__ bias,
                                     float* __restrict__ out,
                                     int n_nodes) {
    const int lane  = threadIdx.x & 31;
    const int wave  = threadIdx.x >> 5;        // 0..7 -> N-tile
    const int m0    = blockIdx.x * 16;
    const int n0    = wave * 16;
    const int lmod  = lane & 15;
    const int khalf = (lane < 16) ? 0 : 2;

    // Bias depends only on N -> same value in all 8 accumulator VGPRs.
    const float bv = bias[n0 + lmod];
    v8f c;
#pragma unroll
    for (int i = 0; i < 8; ++i) c[i] = bv;

    int mload = m0 + lmod;
    if (mload >= n_nodes) mload = n_nodes - 1;   // clamp (keep EXEC all-1s for WMMA)

    const float* arow = agg + (long long)mload * IN_FEATS + khalf;
    const float* brow = W   + (long long)(n0 + lmod) * IN_FEATS + khalf;

#pragma unroll 8
    for (int k0 = 0; k0 < IN_FEATS; k0 += 4) {
        v2f a, b;
        a[0] = arow[k0];  a[1] = arow[k0 + 1];
        b[0] = brow[k0];  b[1] = brow[k0 + 1];
        c = __builtin_amdgcn_wmma_f32_16x16x4_f32(
                /*neg_a=*/false, a, /*neg_b=*/false, b,
                /*c_mod=*/(short)0, c, /*reuse_a=*/false, /*reuse_b=*/false);
    }

    // Store with ReLU. Uniform (scalar) branch: full tiles are branchless.
    const int mbase = m0 + ((lane < 16) ? 0 : 8);
    float* orow = out + (long long)mbase * OUT_FEATS + n0 + lmod;

    if (m0 + 16 <= n_nodes) {
        // Fast path: whole 16-row tile in range -> 8 unguarded b32 stores.
#pragma unroll
        for (int r = 0; r < 8; ++r) {
            orow[(long long)r * OUT_FEATS] = __builtin_fmaxf(c[r], 0.f);
        }
    } else {
        // Tail tile (only if n_nodes % 16 != 0).
#pragma unroll
        for (int r = 0; r < 8; ++r) {
            if (mbase + r < n_nodes) {
                orow[(long long)r * OUT_FEATS] = __builtin_fmaxf(c[r], 0.f);
            }
        }
    }
}

// ---------------------------------------------------------------------------
// Host launcher. Inputs (setup_inputs order):
//   d_in[0] feature  f32 [N,128]
//   d_in[1] edge_src i64 [E]
//   d_in[2] edge_dst i64 [E]
//   d_in[3] W        f32 [128,128]
//   d_in[4] b        f32 [128]
// d_out: f32 [N,128].  d_ws: agg buffer (N*128 f32 = 51.2 MB).
// ---------------------------------------------------------------------------
extern "C" void kernel_launch(void* const* d_in, const int* in_sizes, int n_in,
                              void* d_out, int out_size, void* d_ws, size_t ws_size,
                              hipStream_t stream) {
    const float*     feature = (const float*)d_in[0];
    const long long* esrc    = (const long long*)d_in[1];
    const long long* edst    = (const long long*)d_in[2];
    const float*     W       = (const float*)d_in[3];
    const float*     bias    = (const float*)d_in[4];
    float*           out     = (float*)d_out;
    float*           agg     = (float*)d_ws;

    const long long n_nodes = in_sizes[0] / IN_FEATS;
    const long long n_edges = in_sizes[1];

    // 1) zero agg
    {
        long long n4 = n_nodes * IN_FEATS / 4;
        int blocks = (int)((n4 + 255) / 256);
        gcn_zero_kernel<<<blocks, 256, 0, stream>>>((float4*)agg, n4);
    }

    // 2) aggregate: one wave per edge, 8 waves per 256-thread block
    {
        long long waves_needed = n_edges;
        int blocks = (int)((waves_needed + 7) / 8);
        gcn_aggregate_kernel<<<blocks, 256, 0, stream>>>(feature, esrc, edst, agg, n_edges);
    }

    // 3) WMMA GEMM + bias + relu: one block per 16 nodes
    {
        int blocks = (int)((n_nodes + 15) / 16);
        gcn_gemm_relu_kernel<<<blocks, 256, 0, stream>>>(agg, W, bias, out, (int)n_nodes);
    }
}